// SpecFusedMoE_52742198395537
// MI455X (gfx1250) — compile-verified
//
#include <hip/hip_runtime.h>

typedef __attribute__((ext_vector_type(2))) float        v2f;
typedef __attribute__((ext_vector_type(8))) float        v8f;
typedef __attribute__((ext_vector_type(4))) unsigned int v4u;
typedef __attribute__((ext_vector_type(4))) int          v4i;
typedef __attribute__((ext_vector_type(8))) int          v8i;

#define B_TOK 16
#define E_EXP 64
#define H_DIM 2048
#define I_DIM 768
#define TOPK  8
#define EG    4

#define KCHUNK 32          // K elements per TDM chunk (dim0 of the 2D tile)
#define PITCH  36          // LDS row pitch in floats (32 data + 4 pad dwords)
// D# group1 word0: data_size=2 (4B) <<16 | pad_enable<<20 | pad_interval=4 (32 dw) <<22
//                  | pad_amount=3 (4 dwords) <<25
#define TDM_FLAGS 0x07120000u

#if defined(__HIP_DEVICE_COMPILE__) && __has_builtin(__builtin_amdgcn_tensor_load_to_lds)
#define USE_TDM 1
#else
#define USE_TDM 0
#endif

__device__ __forceinline__ v8f wmma4(v2f a, v2f b, v8f c) {
    return __builtin_amdgcn_wmma_f32_16x16x4_f32(false, a, false, b,
                                                 (short)0, c, false, false);
}

#if USE_TDM
// Issue one TDM 2D-tile load: tile_rows rows of tile_k fp32 elements,
// row stride row_stride (elements), into LDS at lds_byte_off (padded per TDM_FLAGS).
__device__ __forceinline__ void tdm_load_tile(unsigned lds_byte_off,
                                              const float* gaddr,
                                              unsigned tile_k, unsigned tile_rows,
                                              unsigned tensor_d0, unsigned tensor_d1,
                                              unsigned long long row_stride) {
    unsigned long long ga = (unsigned long long)(uintptr_t)gaddr;
    v4u g0;
    g0.x = 1u;                                                    // count=1 (user D#)
    g0.y = lds_byte_off;                                          // lds_addr
    g0.z = (unsigned)ga;                                          // global_addr[31:0]
    g0.w = (unsigned)((ga >> 32) & 0x01FFFFFFu) | 0x80000000u;    // addr[56:32] | type=2

    v8i g1;
    g1[0] = (int)TDM_FLAGS;                                       // mask=0, 4B, padding
    g1[1] = (int)((tensor_d0 & 0xFFFFu) << 16);                   // tensor_dim0[15:0]
    g1[2] = (int)(((tensor_d0 >> 16) & 0xFFFFu) |
                  ((tensor_d1 & 0xFFFFu) << 16));                 // d0[31:16] | d1[15:0]
    g1[3] = (int)(((tensor_d1 >> 16) & 0xFFFFu) |
                  ((tile_k & 0xFFFFu) << 16));                    // d1[31:16] | tile_dim0
    g1[4] = (int)(tile_rows & 0xFFFFu);                           // tile_dim1, tile_dim2=0
    g1[5] = (int)(unsigned)(row_stride & 0xFFFFFFFFull);          // dim0_stride[31:0]
    g1[6] = (int)(unsigned)((row_stride >> 32) & 0xFFFFull);      // dim0_stride[47:32]
    g1[7] = 0;                                                    // dim1_stride hi (unused)

    v4i z4 = {0, 0, 0, 0};
#if defined(__clang_major__) && (__clang_major__ >= 23)
    v8i z8 = {0, 0, 0, 0, 0, 0, 0, 0};
    __builtin_amdgcn_tensor_load_to_lds(g0, g1, z4, z4, z8, 0);
#else
    __builtin_amdgcn_tensor_load_to_lds(g0, g1, z4, z4, 0);
#endif
}
#endif // USE_TDM

// ---------------------------------------------------------------------------
// Kernel 1: routing (softmax -> top-8 -> renorm -> dense combine) + zero out.
// ---------------------------------------------------------------------------
__global__ void moe_route(const float* __restrict__ logits,   // [16][64]
                          float* __restrict__ combine,        // [16][64]
                          int* __restrict__ active,           // [64]
                          float* __restrict__ out)            // [16][2048]
{
    const int t = threadIdx.x;           // 64 threads
    active[t] = 0;
    for (int i = t; i < B_TOK * E_EXP; i += 64) combine[i] = 0.0f;
    for (int i = t; i < B_TOK * H_DIM; i += 64) out[i] = 0.0f;
    __syncthreads();

    if (t < B_TOK) {
        float p[E_EXP];
        const float* row = logits + t * E_EXP;
        float mx = row[0];
        for (int e = 1; e < E_EXP; ++e) mx = fmaxf(mx, row[e]);
        float sum = 0.0f;
        for (int e = 0; e < E_EXP; ++e) { p[e] = __expf(row[e] - mx); sum += p[e]; }
        const float inv = 1.0f / sum;
        for (int e = 0; e < E_EXP; ++e) p[e] *= inv;

        int   idx[TOPK];
        float w[TOPK];
        float wsum = 0.0f;
        for (int k = 0; k < TOPK; ++k) {
            int bi = 0; float bv = -1.0f;
            for (int e = 0; e < E_EXP; ++e)
                if (p[e] > bv) { bv = p[e]; bi = e; }
            idx[k] = bi; w[k] = bv; wsum += bv;
            p[bi] = -2.0f;
        }
        const float winv = 1.0f / wsum;
        for (int k = 0; k < TOPK; ++k) {
            combine[t * E_EXP + idx[k]] = w[k] * winv;
            atomicOr(&active[idx[k]], 1);
        }
    }
}

// ---------------------------------------------------------------------------
// Kernel 2: act[e][b][i] = silu(hidden @ gate_w[e]^T) * (hidden @ up_w[e]^T)
// Grid (E, I/64), 128 threads = 4 waves; wave owns a 16-wide I tile.
// TDM path: wave 0 DMAs gate/up [64 rows x 32 K] tiles into padded LDS,
// double-buffered on TENSORcnt; all waves feed v_wmma from ds_load_b64.
// ---------------------------------------------------------------------------
__global__ void __launch_bounds__(128)
moe_gate_up(const float* __restrict__ hidden,   // [16][2048]
            const float* __restrict__ gate_w,   // [64][768][2048]
            const float* __restrict__ up_w,     // [64][768][2048]
            const int*   __restrict__ active,   // [64]
            float* __restrict__ act)            // [64][16][768]
{
    const int e = blockIdx.x;
    if (!active[e]) return;                     // block-uniform: EXEC stays all-1s

    const int wave  = threadIdx.x >> 5;
    const int lane  = threadIdx.x & 31;
    const int row   = lane & 15;                // A: token row / B: N column (i row)
    const int hi    = lane >> 4;                // K sub-pair select
    const int ib0   = blockIdx.y * 64;          // block's first i row
    const int ibase = ib0 + wave * 16;

    const float* Ap = hidden + (size_t)row * H_DIM + 2 * hi;
    v8f accg = {};
    v8f accu = {};

#if USE_TDM
    __shared__ float lds[4 * 64 * PITCH];       // [gate0, up0, gate1, up1]
    const unsigned seg = 64u * PITCH * 4u;      // bytes per segment
    const unsigned lds0 = (unsigned)(uintptr_t)(void*)&lds[0];
    const float* gbase = gate_w + ((size_t)e * I_DIM + ib0) * H_DIM;
    const float* ubase = up_w   + ((size_t)e * I_DIM + ib0) * H_DIM;

    if (wave == 0) {
        tdm_load_tile(lds0 + 0u * seg, gbase, KCHUNK, 64, H_DIM, I_DIM, H_DIM);
        tdm_load_tile(lds0 + 1u * seg, ubase, KCHUNK, 64, H_DIM, I_DIM, H_DIM);
    }

    const int nchunk = H_DIM / KCHUNK;          // 64
    for (int c = 0; c < nchunk; ++c) {
        const int cur = c & 1;
        if (wave == 0) {
            if (c + 1 < nchunk) {
                const unsigned nxt = 1u - (unsigned)cur;
                tdm_load_tile(lds0 + (2u * nxt + 0u) * seg,
                              gbase + (size_t)(c + 1) * KCHUNK,
                              KCHUNK, 64, H_DIM, I_DIM, H_DIM);
                tdm_load_tile(lds0 + (2u * nxt + 1u) * seg,
                              ubase + (size_t)(c + 1) * KCHUNK,
                              KCHUNK, 64, H_DIM, I_DIM, H_DIM);
                __builtin_amdgcn_s_wait_tensorcnt(2);   // chunk c landed
            } else {
                __builtin_amdgcn_s_wait_tensorcnt(0);
            }
        }
        __syncthreads();                         // publish LDS tile to all waves

        const float* gl = lds + (size_t)(2 * cur + 0) * 64 * PITCH
                              + (size_t)(wave * 16 + row) * PITCH + 2 * hi;
        const float* ul = lds + (size_t)(2 * cur + 1) * 64 * PITCH
                              + (size_t)(wave * 16 + row) * PITCH + 2 * hi;
        const float* ag = Ap + (size_t)c * KCHUNK;
#pragma unroll
        for (int k0 = 0; k0 < KCHUNK; k0 += 4) {
            v2f a  = *(const v2f*)(ag + k0);
            v2f bg = *(const v2f*)(gl + k0);
            v2f bu = *(const v2f*)(ul + k0);
            accg = wmma4(a, bg, accg);
            accu = wmma4(a, bu, accu);
        }
        __syncthreads();                         // tile fully consumed before reuse
    }
#else
    const float* Gp = gate_w + ((size_t)e * I_DIM + ibase + row) * H_DIM + 2 * hi;
    const float* Up = up_w   + ((size_t)e * I_DIM + ibase + row) * H_DIM + 2 * hi;
#pragma unroll 4
    for (int k0 = 0; k0 < H_DIM; k0 += 4) {
        v2f a  = *(const v2f*)(Ap + k0);
        v2f bg = *(const v2f*)(Gp + k0);
        v2f bu = *(const v2f*)(Up + k0);
        accg = wmma4(a, bg, accg);
        accu = wmma4(a, bu, accu);
    }
#endif

    // D layout: reg r, lane -> token m = r + 8*hi, col n = row (i index)
    float* dst = act + ((size_t)e * B_TOK + 8 * hi) * I_DIM + ibase + row;
#pragma unroll
    for (int r = 0; r < 8; ++r) {
        const float g = accg[r];
        const float u = accu[r];
        const float s = g / (1.0f + __expf(-g));    // silu
        dst[(size_t)r * I_DIM] = s * u;
    }
}

// ---------------------------------------------------------------------------
// Kernel 3: out[b][h] += sum_e combine[b][e] * (act[e] @ down_w[e]^T)
// Grid (H/128, E/EG), 256 threads = 8 waves; wave owns a 16-wide H tile,
// reduces EG experts locally, then global_atomic_add_f32 the tile.
// TDM path: wave 0 DMAs down_w [128 rows x 32 K] tiles into padded LDS,
// double-buffered on TENSORcnt (this is the dominant 402 MB HBM stream).
// ---------------------------------------------------------------------------
__global__ void __launch_bounds__(256)
moe_down(const float* __restrict__ act,      // [64][16][768]
         const float* __restrict__ down_w,   // [64][2048][768]
         const float* __restrict__ combine,  // [16][64]
         const int*   __restrict__ active,   // [64]
         float* __restrict__ out)            // [16][2048]
{
    const int wave  = threadIdx.x >> 5;
    const int lane  = threadIdx.x & 31;
    const int row   = lane & 15;
    const int hi    = lane >> 4;
    const int hb0   = blockIdx.x * 128;         // block's first h row
    const int hbase = hb0 + wave * 16;
    const int e0    = blockIdx.y * EG;

#if USE_TDM
    __shared__ float lds[2 * 128 * PITCH];      // double-buffered 128x32 tile
    const unsigned seg  = 128u * PITCH * 4u;    // bytes per buffer
    const unsigned lds0 = (unsigned)(uintptr_t)(void*)&lds[0];
#endif

    v8f outacc = {};
    for (int eo = 0; eo < EG; ++eo) {
        const int e = e0 + eo;
        if (!active[e]) continue;               // block-uniform

        const float* Ap = act + ((size_t)e * B_TOK + row) * I_DIM + 2 * hi;
        v8f acc = {};

#if USE_TDM
        const float* dbase = down_w + ((size_t)e * H_DIM + hb0) * I_DIM;
        if (wave == 0)
            tdm_load_tile(lds0, dbase, KCHUNK, 128, I_DIM, H_DIM, I_DIM);

        const int nchunk = I_DIM / KCHUNK;      // 24
        for (int c = 0; c < nchunk; ++c) {
            const int cur = c & 1;
            if (wave == 0) {
                if (c + 1 < nchunk) {
                    tdm_load_tile(lds0 + (1u - (unsigned)cur) * seg,
                                  dbase + (size_t)(c + 1) * KCHUNK,
                                  KCHUNK, 128, I_DIM, H_DIM, I_DIM);
                    __builtin_amdgcn_s_wait_tensorcnt(1);   // chunk c landed
                } else {
                    __builtin_amdgcn_s_wait_tensorcnt(0);
                }
            }
            __syncthreads();                     // publish LDS tile to all waves

            const float* bl = lds + (size_t)cur * 128 * PITCH
                                  + (size_t)(wave * 16 + row) * PITCH + 2 * hi;
            const float* ap = Ap + (size_t)c * KCHUNK;
#pragma unroll
            for (int k0 = 0; k0 < KCHUNK; k0 += 4) {
                v2f a = *(const v2f*)(ap + k0);
                v2f b = *(const v2f*)(bl + k0);
                acc = wmma4(a, b, acc);
            }
            __syncthreads();                     // tile fully consumed before reuse
        }
#else
        const float* Bp = down_w + ((size_t)e * H_DIM + hbase + row) * I_DIM + 2 * hi;
#pragma unroll 4
        for (int k0 = 0; k0 < I_DIM; k0 += 4) {
            v2f a = *(const v2f*)(Ap + k0);
            v2f b = *(const v2f*)(Bp + k0);
            acc = wmma4(a, b, acc);
        }
#endif

#pragma unroll
        for (int r = 0; r < 8; ++r) {
            const int m = r + 8 * hi;           // token row
            outacc[r] += combine[m * E_EXP + e] * acc[r];
        }
    }

    float* dst = out + (size_t)(8 * hi) * H_DIM + hbase + row;
#pragma unroll
    for (int r = 0; r < 8; ++r)
        atomicAdd(dst + (size_t)r * H_DIM, outacc[r]);
}

// ---------------------------------------------------------------------------
extern "C" void kernel_launch(void* const* d_in, const int* in_sizes, int n_in,
                              void* d_out, int out_size, void* d_ws, size_t ws_size,
                              hipStream_t stream) {
    (void)in_sizes; (void)n_in; (void)out_size; (void)ws_size;
    const float* hidden = (const float*)d_in[0];
    const float* logits = (const float*)d_in[1];
    const float* gate_w = (const float*)d_in[2];
    const float* up_w   = (const float*)d_in[3];
    const float* down_w = (const float*)d_in[4];
    float* out = (float*)d_out;

    char* ws = (char*)d_ws;
    const size_t act_bytes = (size_t)E_EXP * B_TOK * I_DIM * sizeof(float); // 3.0 MB
    float* act     = (float*)ws;
    float* combine = (float*)(ws + act_bytes);
    int*   active  = (int*)(ws + act_bytes + (size_t)B_TOK * E_EXP * sizeof(float));

    moe_route<<<1, 64, 0, stream>>>(logits, combine, active, out);
    moe_gate_up<<<dim3(E_EXP, I_DIM / 64), 128, 0, stream>>>(hidden, gate_w, up_w,
                                                             active, act);
    moe_down<<<dim3(H_DIM / 128, E_EXP / EG), 256, 0, stream>>>(act, down_w, combine,
                                                                active, out);
}